// myGIN_26182120636972
// MI455X (gfx1250) — compile-verified
//
#include <hip/hip_runtime.h>
#include <hip/hip_bf16.h>

#define N_NODES 50000
#define N_EDGES 800000
#define IN_C 100
#define HID_C 128
#define OUT_C 40
#define OUT_PAD 48

typedef float v2f __attribute__((ext_vector_type(2)));
typedef float v8f __attribute__((ext_vector_type(8)));

// ---------------------------------------------------------------------------
// agg1 = x  (element-wise copy, fully overwrites workspace each call)
// ---------------------------------------------------------------------------
__global__ __launch_bounds__(256) void gin_init_agg1(const float* __restrict__ x,
                                                     float* __restrict__ agg1) {
    unsigned idx = blockIdx.x * 256u + threadIdx.x;
    const unsigned total = N_NODES * IN_C;
    if (idx < total) agg1[idx] = x[idx];
}

// ---------------------------------------------------------------------------
// agg1[dst[e], c] += x[src[e], c]   (GIN segment-sum, layer 1)
// ---------------------------------------------------------------------------
__global__ __launch_bounds__(256) void gin_scatter1(const float* __restrict__ x,
                                                    const int* __restrict__ src,
                                                    const int* __restrict__ dst,
                                                    float* __restrict__ agg1) {
    long long idx = (long long)blockIdx.x * 256 + threadIdx.x;
    const long long total = (long long)N_EDGES * IN_C;
    if (idx >= total) return;
    unsigned e = (unsigned)(idx / IN_C);
    unsigned c = (unsigned)(idx - (long long)e * IN_C);
    atomicAdd(&agg1[(size_t)dst[e] * IN_C + c], x[(size_t)src[e] * IN_C + c]);
}

// ---------------------------------------------------------------------------
// agg2[dst[e], c] += h1[src[e], c]  (GIN segment-sum, layer 2; C=128)
// ---------------------------------------------------------------------------
__global__ __launch_bounds__(256) void gin_scatter2(const float* __restrict__ h1,
                                                    const int* __restrict__ src,
                                                    const int* __restrict__ dst,
                                                    float* __restrict__ agg2) {
    long long idx = (long long)blockIdx.x * 256 + threadIdx.x;
    const long long total = (long long)N_EDGES * HID_C;
    if (idx >= total) return;
    unsigned e = (unsigned)(idx >> 7);
    unsigned c = (unsigned)(idx & 127);
    atomicAdd(&agg2[(size_t)dst[e] * HID_C + c], h1[(size_t)src[e] * HID_C + c]);
}

// ---------------------------------------------------------------------------
// Layer-1 MLP: h = relu( relu(agg1 @ W1a + b1a) @ W1b + b1b )
// One block = 16 rows, 8 waves (each wave owns one 16-wide column tile).
// Writes h1 and also agg2 (pre-initializing the layer-2 aggregation buffer).
// ---------------------------------------------------------------------------
__global__ __launch_bounds__(256) void gin_mlp1(const float* __restrict__ agg1,
                                                const float* __restrict__ W1a,
                                                const float* __restrict__ b1a,
                                                const float* __restrict__ W1b,
                                                const float* __restrict__ b1b,
                                                float* __restrict__ h1,
                                                float* __restrict__ agg2) {
    __shared__ float sA[16][IN_C];   // aggregated inputs (16 x 100)
    __shared__ float sH[16][HID_C];  // stage-1 intermediate (16 x 128)

    const int row0 = blockIdx.x * 16;
    const int tid  = threadIdx.x;

    for (int i = tid; i < 16 * IN_C; i += 256) {
        int r = i / IN_C, c = i - r * IN_C;
        sA[r][c] = agg1[(size_t)(row0 + r) * IN_C + c];
    }
    __syncthreads();

    const int wave = tid >> 5;        // 0..7 -> column tile
    const int lane = tid & 31;
    const int half = lane >> 4;       // 0/1
    const int lrow = lane & 15;       // A row / B column within tile
    const int n    = wave * 16 + lrow;

    // ---- stage 1: [16x100] @ [100x128], K-step 4 -------------------------
    v8f acc = {0.f, 0.f, 0.f, 0.f, 0.f, 0.f, 0.f, 0.f};
    for (int k0 = 0; k0 < IN_C; k0 += 4) {
        const int k = k0 + 2 * half;
        v2f a, b;
        a[0] = sA[lrow][k];
        a[1] = sA[lrow][k + 1];
        b[0] = W1a[(size_t)k * HID_C + n];
        b[1] = W1a[(size_t)(k + 1) * HID_C + n];
        acc = __builtin_amdgcn_wmma_f32_16x16x4_f32(false, a, false, b,
                                                    (short)0, acc, false, false);
    }
    {
        const float bias = b1a[n];
#pragma unroll
        for (int i = 0; i < 8; ++i) {
            int m = i + 8 * half;
            float v = acc[i] + bias;
            sH[m][n] = v > 0.f ? v : 0.f;
        }
    }
    __syncthreads();

    // ---- stage 2: [16x128] @ [128x128] -----------------------------------
    v8f acc2 = {0.f, 0.f, 0.f, 0.f, 0.f, 0.f, 0.f, 0.f};
    for (int k0 = 0; k0 < HID_C; k0 += 4) {
        const int k = k0 + 2 * half;
        v2f a, b;
        a[0] = sH[lrow][k];
        a[1] = sH[lrow][k + 1];
        b[0] = W1b[(size_t)k * HID_C + n];
        b[1] = W1b[(size_t)(k + 1) * HID_C + n];
        acc2 = __builtin_amdgcn_wmma_f32_16x16x4_f32(false, a, false, b,
                                                     (short)0, acc2, false, false);
    }
    {
        const float bias = b1b[n];
#pragma unroll
        for (int i = 0; i < 8; ++i) {
            int m = i + 8 * half;
            float v = acc2[i] + bias;
            v = v > 0.f ? v : 0.f;   // trailing ReLU of layer 1
            size_t off = (size_t)(row0 + m) * HID_C + n;
            h1[off]   = v;
            agg2[off] = v;           // agg2 starts as a copy of h1
        }
    }
}

// ---------------------------------------------------------------------------
// Layer-2 MLP + log_softmax:
//   logits = relu(agg2 @ W2a + b2a) @ W2b + b2b ; out = log_softmax(logits)
// One block = 16 rows, 3 waves (48 padded columns; n >= 40 masked to zero).
// ---------------------------------------------------------------------------
__global__ __launch_bounds__(96) void gin_mlp2(const float* __restrict__ agg2,
                                               const float* __restrict__ W2a,
                                               const float* __restrict__ b2a,
                                               const float* __restrict__ W2b,
                                               const float* __restrict__ b2b,
                                               float* __restrict__ out) {
    __shared__ float sA[16][HID_C];     // aggregated hidden (16 x 128)
    __shared__ float sH[16][OUT_PAD];   // stage-1 intermediate (padded)
    __shared__ float sL[16][OUT_PAD];   // logits (padded)
    __shared__ float sLse[16];

    const int row0 = blockIdx.x * 16;
    const int tid  = threadIdx.x;

    for (int i = tid; i < 16 * HID_C; i += 96) {
        int r = i >> 7, c = i & 127;
        sA[r][c] = agg2[(size_t)(row0 + r) * HID_C + c];
    }
    __syncthreads();

    const int wave = tid >> 5;        // 0..2 -> column tile
    const int lane = tid & 31;
    const int half = lane >> 4;
    const int lrow = lane & 15;
    const int n    = wave * 16 + lrow;          // 0..47
    const bool nvalid = (n < OUT_C);

    // ---- stage 1: [16x128] @ [128x40] (padded to 48 cols) ----------------
    v8f acc = {0.f, 0.f, 0.f, 0.f, 0.f, 0.f, 0.f, 0.f};
    for (int k0 = 0; k0 < HID_C; k0 += 4) {
        const int k = k0 + 2 * half;
        v2f a, b;
        a[0] = sA[lrow][k];
        a[1] = sA[lrow][k + 1];
        b[0] = nvalid ? W2a[(size_t)k * OUT_C + n] : 0.f;
        b[1] = nvalid ? W2a[(size_t)(k + 1) * OUT_C + n] : 0.f;
        acc = __builtin_amdgcn_wmma_f32_16x16x4_f32(false, a, false, b,
                                                    (short)0, acc, false, false);
    }
    {
        const float bias = nvalid ? b2a[n] : 0.f;
#pragma unroll
        for (int i = 0; i < 8; ++i) {
            int m = i + 8 * half;
            float v = acc[i] + bias;
            sH[m][n] = v > 0.f ? v : 0.f;
        }
    }
    __syncthreads();

    // ---- stage 2: [16x40] @ [40x40] (K=40, 10 WMMA steps) ----------------
    v8f acc2 = {0.f, 0.f, 0.f, 0.f, 0.f, 0.f, 0.f, 0.f};
    for (int k0 = 0; k0 < OUT_C; k0 += 4) {
        const int k = k0 + 2 * half;
        v2f a, b;
        a[0] = sH[lrow][k];
        a[1] = sH[lrow][k + 1];
        b[0] = nvalid ? W2b[(size_t)k * OUT_C + n] : 0.f;
        b[1] = nvalid ? W2b[(size_t)(k + 1) * OUT_C + n] : 0.f;
        acc2 = __builtin_amdgcn_wmma_f32_16x16x4_f32(false, a, false, b,
                                                     (short)0, acc2, false, false);
    }
    {
        const float bias = nvalid ? b2b[n] : 0.f;
#pragma unroll
        for (int i = 0; i < 8; ++i) {
            int m = i + 8 * half;
            sL[m][n] = acc2[i] + bias;
        }
    }
    __syncthreads();

    // ---- log_softmax over 40 logits per row ------------------------------
    if (tid < 16) {
        float mx = -3.402823466e38f;
        for (int c = 0; c < OUT_C; ++c) mx = fmaxf(mx, sL[tid][c]);
        float s = 0.f;
        for (int c = 0; c < OUT_C; ++c) s += expf(sL[tid][c] - mx);
        sLse[tid] = mx + logf(s);
    }
    __syncthreads();
    for (int i = tid; i < 16 * OUT_C; i += 96) {
        int r = i / OUT_C, c = i - r * OUT_C;
        out[(size_t)(row0 + r) * OUT_C + c] = sL[r][c] - sLse[r];
    }
}

// ---------------------------------------------------------------------------
extern "C" void kernel_launch(void* const* d_in, const int* in_sizes, int n_in,
                              void* d_out, int out_size, void* d_ws, size_t ws_size,
                              hipStream_t stream) {
    const float* x    = (const float*)d_in[0];
    const int*   ei   = (const int*)d_in[1];        // [2, N_EDGES]
    const float* W1a  = (const float*)d_in[2];
    const float* b1a  = (const float*)d_in[3];
    const float* W1b  = (const float*)d_in[4];
    const float* b1b  = (const float*)d_in[5];
    const float* W2a  = (const float*)d_in[6];
    const float* b2a  = (const float*)d_in[7];
    const float* W2b  = (const float*)d_in[8];
    const float* b2b  = (const float*)d_in[9];
    float* out = (float*)d_out;

    const int* src = ei;
    const int* dst = ei + N_EDGES;

    // workspace layout (all fully overwritten every call before accumulation)
    float* agg1 = (float*)d_ws;                     // [N, 100]  20.0 MB
    float* h1   = agg1 + (size_t)N_NODES * IN_C;    // [N, 128]  25.6 MB
    float* agg2 = h1   + (size_t)N_NODES * HID_C;   // [N, 128]  25.6 MB

    // 1) agg1 = x
    {
        unsigned total = N_NODES * IN_C;
        gin_init_agg1<<<(total + 255) / 256, 256, 0, stream>>>(x, agg1);
    }
    // 2) agg1 += segment_sum(x[src], dst)
    {
        long long total = (long long)N_EDGES * IN_C;
        gin_scatter1<<<(unsigned)((total + 255) / 256), 256, 0, stream>>>(x, src, dst, agg1);
    }
    // 3) h1 = MLP1(agg1); agg2 = h1
    gin_mlp1<<<N_NODES / 16, 256, 0, stream>>>(agg1, W1a, b1a, W1b, b1b, h1, agg2);
    // 4) agg2 += segment_sum(h1[src], dst)
    {
        long long total = (long long)N_EDGES * HID_C;
        gin_scatter2<<<(unsigned)((total + 255) / 256), 256, 0, stream>>>(h1, src, dst, agg2);
    }
    // 5) out = log_softmax(MLP2(agg2))
    gin_mlp2<<<N_NODES / 16, 96, 0, stream>>>(agg2, W2a, b2a, W2b, b2b, out);
}